// Attention_12077448036617
// MI455X (gfx1250) — compile-verified
//
#include <hip/hip_runtime.h>
#include <math.h>

typedef __attribute__((ext_vector_type(2))) float v2f;
typedef __attribute__((ext_vector_type(8))) float v8f;

#define B_   64
#define T_   2048
#define EMB_ 512
#define QDIM_ 1024
#define ADIM_ 128
#define LCH_ 32
#define KS_  31
#define PAD_ 15
#define TT   128     // t-tile per energies block
#define TCH  128     // t-chunk per context-partial block
#define NCH  (T_ / TCH)

// ---------------------------------------------------------------------------
// Kernel 1: processed_query = query @ W_query^T   (64x1024 @ 1024x128)
// Pure f32 WMMA 16x16x4. 32 waves total = 4 blocks x 8 waves.
// ---------------------------------------------------------------------------
__global__ __launch_bounds__(256) void pq_kernel(
    const float* __restrict__ query,    // (B,QDIM)
    const float* __restrict__ W_query,  // (ADIM,QDIM)
    float* __restrict__ pq)             // (B,ADIM)
{
    const int gw    = blockIdx.x * 8 + (threadIdx.x >> 5);  // 0..31
    const int lane  = threadIdx.x & 31;
    const int lhalf = lane & 15;
    const int khi   = (lane >> 4) * 2;       // 0 for lanes 0-15, 2 for 16-31
    const int m0    = (gw >> 3) * 16;        // batch tile
    const int n0    = (gw & 7) * 16;         // a tile

    v8f d = {};
    for (int k0 = 0; k0 < QDIM_; k0 += 4) {
        v2f a, b;
        // A: 16x4 f32 layout (ISA 7.12.2): lanes0-15 K=k0,k0+1 ; lanes16-31 K=k0+2,k0+3
        a.x = query[(m0 + lhalf) * QDIM_ + k0 + khi + 0];
        a.y = query[(m0 + lhalf) * QDIM_ + k0 + khi + 1];
        // B[k][n] = W_query[n][k]
        b.x = W_query[(n0 + lhalf) * QDIM_ + k0 + khi + 0];
        b.y = W_query[(n0 + lhalf) * QDIM_ + k0 + khi + 1];
        d = __builtin_amdgcn_wmma_f32_16x16x4_f32(false, a, false, b,
                                                  (short)0, d, false, false);
    }
    const int tsel = (lane < 16) ? 0 : 8;    // D: VGPR r -> M=r (lo half) / r+8 (hi half)
#pragma unroll
    for (int r = 0; r < 8; ++r)
        pq[(m0 + r + tsel) * ADIM_ + n0 + lhalf] = d[r];
}

// ---------------------------------------------------------------------------
// Kernel 2 (hot): fused conv1d + loc-projection (WMMA) + tanh energies.
// One block per (b, 128-t tile).  Streams processed_memory exactly once.
// ---------------------------------------------------------------------------
__global__ __launch_bounds__(256) void energies_kernel(
    const float* __restrict__ awc,     // (B,2,T)
    const float* __restrict__ pm,      // (B,T,ADIM)
    const float* __restrict__ conv_w,  // (LCH,2,KS)
    const float* __restrict__ conv_b,  // (LCH)
    const float* __restrict__ W_loc,   // (ADIM,LCH)
    const float* __restrict__ W_v,     // (ADIM)
    const float* __restrict__ pq,      // (B,ADIM)
    float* __restrict__ energies)      // (B,T)
{
    __shared__ float s_awc[2][TT + KS_ - 1];   // padded input rows
    __shared__ float s_cw[LCH_ * 2 * KS_];     // conv weights
    __shared__ float s_cb[LCH_];
    __shared__ float s_conv[LCH_][TT];         // conv output [c][t]
    __shared__ float s_wlocT[LCH_ * ADIM_];    // W_loc transposed [c][a] (bank-friendly)
    __shared__ float s_pq[ADIM_];
    __shared__ float s_wv[ADIM_];

    const int b   = blockIdx.y;
    const int t0  = blockIdx.x * TT;
    const int tid = threadIdx.x;

    for (int i = tid; i < LCH_ * 2 * KS_; i += 256) s_cw[i] = conv_w[i];
    if (tid < LCH_) s_cb[tid] = conv_b[tid];
    for (int i = tid; i < LCH_ * ADIM_; i += 256) {
        int c = i / ADIM_, a = i % ADIM_;
        s_wlocT[i] = W_loc[a * LCH_ + c];
    }
    if (tid < ADIM_) { s_pq[tid] = pq[b * ADIM_ + tid]; s_wv[tid] = W_v[tid]; }
    for (int i = tid; i < 2 * (TT + KS_ - 1); i += 256) {
        int row = i / (TT + KS_ - 1), j = i % (TT + KS_ - 1);
        int tg = t0 - PAD_ + j;
        s_awc[row][j] = (tg >= 0 && tg < T_) ? awc[(b * 2 + row) * T_ + tg] : 0.0f;
    }
    __syncthreads();

    // conv1d(2->32, k=31, pad=15) into LDS
    for (int i = tid; i < LCH_ * TT; i += 256) {
        int c = i / TT, tl = i % TT;
        float acc = s_cb[c];
        const float* w0 = &s_cw[(c * 2 + 0) * KS_];
        const float* w1 = &s_cw[(c * 2 + 1) * KS_];
#pragma unroll
        for (int k = 0; k < KS_; ++k)
            acc = fmaf(s_awc[0][tl + k], w0[k], fmaf(s_awc[1][tl + k], w1[k], acc));
        s_conv[c][tl] = acc;
    }
    __syncthreads();

    // Each wave owns a 16-row t-tile; loop the 8 a-tiles.
    // pa(16t x 16a) = conv(16t x 32c) @ wlocT(32c x 16a)  via 8x WMMA f32 16x16x4,
    // then fuse +pq +pm, tanh, *W_v, reduce over a.
    const int wave  = tid >> 5;
    const int lane  = tid & 31;
    const int lhalf = lane & 15;
    const int khi   = (lane >> 4) * 2;
    const int m0    = wave * 16;
    const int tsel  = (lane < 16) ? 0 : 8;

    float e_acc[8];
#pragma unroll
    for (int r = 0; r < 8; ++r) e_acc[r] = 0.0f;

    for (int n0 = 0; n0 < ADIM_; n0 += 16) {
        v8f d = {};
#pragma unroll
        for (int k0 = 0; k0 < LCH_; k0 += 4) {
            v2f a, bm;
            a.x  = s_conv[k0 + khi + 0][m0 + lhalf];
            a.y  = s_conv[k0 + khi + 1][m0 + lhalf];
            bm.x = s_wlocT[(k0 + khi + 0) * ADIM_ + n0 + lhalf];
            bm.y = s_wlocT[(k0 + khi + 1) * ADIM_ + n0 + lhalf];
            d = __builtin_amdgcn_wmma_f32_16x16x4_f32(false, a, false, bm,
                                                      (short)0, d, false, false);
        }
        const int   aidx = n0 + lhalf;
        const float pqv  = s_pq[aidx];
        const float wv   = s_wv[aidx];
        const float* pmb = pm + ((long)b * T_ + t0 + m0 + tsel) * ADIM_ + aidx;
#pragma unroll
        for (int r = 0; r < 8; ++r) {
            float x = d[r] + pqv + pmb[(long)r * ADIM_];
            e_acc[r] += wv * tanhf(x);
        }
    }

    // reduce each e_acc[r] across its 16-lane half; lanes 0/16 write t / t+8.
#pragma unroll
    for (int r = 0; r < 8; ++r) {
        float v = e_acc[r];
        v += __shfl_xor(v, 8, 32);
        v += __shfl_xor(v, 4, 32);
        v += __shfl_xor(v, 2, 32);
        v += __shfl_xor(v, 1, 32);
        if (lhalf == 0)
            energies[(long)b * T_ + t0 + m0 + r + tsel] = v;
    }
}

// ---------------------------------------------------------------------------
// Kernel 3: softmax over T per batch row.
// ---------------------------------------------------------------------------
__global__ __launch_bounds__(256) void softmax_kernel(
    const float* __restrict__ energies, float* __restrict__ aw)
{
    const int b = blockIdx.x;
    const int tid = threadIdx.x;
    __shared__ float red[256];

    float m = -INFINITY;
    for (int t = tid; t < T_; t += 256) m = fmaxf(m, energies[b * T_ + t]);
    red[tid] = m; __syncthreads();
    for (int s = 128; s > 0; s >>= 1) {
        if (tid < s) red[tid] = fmaxf(red[tid], red[tid + s]);
        __syncthreads();
    }
    m = red[0]; __syncthreads();

    float sum = 0.0f;
    for (int t = tid; t < T_; t += 256) sum += expf(energies[b * T_ + t] - m);
    red[tid] = sum; __syncthreads();
    for (int s = 128; s > 0; s >>= 1) {
        if (tid < s) red[tid] += red[tid + s];
        __syncthreads();
    }
    const float inv = 1.0f / red[0];
    for (int t = tid; t < T_; t += 256)
        aw[b * T_ + t] = expf(energies[b * T_ + t] - m) * inv;
}

// ---------------------------------------------------------------------------
// Kernel 4: partial context = sum over a 128-t chunk of w[t]*memory[t][e].
// Streams `memory` (256 MB) exactly once; deterministic (no float atomics).
// ---------------------------------------------------------------------------
__global__ __launch_bounds__(256) void context_partial_kernel(
    const float* __restrict__ aw,      // (B,T)
    const float* __restrict__ memory,  // (B,T,EMB)
    float* __restrict__ partial)       // (B,NCH,EMB)
{
    const int b   = blockIdx.y;
    const int ch  = blockIdx.x;          // 0..NCH-1
    const int tid = threadIdx.x;
    __shared__ float s_w[TCH];

    if (tid < TCH) s_w[tid] = aw[b * T_ + ch * TCH + tid];
    __syncthreads();

    const float* mp = memory + ((long)b * T_ + (long)ch * TCH) * EMB_;
    float acc0 = 0.0f, acc1 = 0.0f;
    for (int t = 0; t < TCH; ++t) {
        if (t + 8 < TCH)  // stream prefetch ahead (global_prefetch_b8)
            __builtin_prefetch(&mp[(long)(t + 8) * EMB_ + tid], 0, 0);
        const float w = s_w[t];
        acc0 = fmaf(w, mp[(long)t * EMB_ + tid], acc0);
        acc1 = fmaf(w, mp[(long)t * EMB_ + tid + 256], acc1);
    }
    partial[((long)b * NCH + ch) * EMB_ + tid]       = acc0;
    partial[((long)b * NCH + ch) * EMB_ + tid + 256] = acc1;
}

// ---------------------------------------------------------------------------
// Kernel 5: reduce partials -> context.
// ---------------------------------------------------------------------------
__global__ __launch_bounds__(256) void context_reduce_kernel(
    const float* __restrict__ partial, float* __restrict__ ctx)
{
    const int b = blockIdx.x;
    for (int e = threadIdx.x; e < EMB_; e += 256) {
        float acc = 0.0f;
        for (int c = 0; c < NCH; ++c)
            acc += partial[((long)b * NCH + c) * EMB_ + e];
        ctx[b * EMB_ + e] = acc;
    }
}

// ---------------------------------------------------------------------------
extern "C" void kernel_launch(void* const* d_in, const int* in_sizes, int n_in,
                              void* d_out, int out_size, void* d_ws, size_t ws_size,
                              hipStream_t stream) {
    const float* query  = (const float*)d_in[0];  // (64,1024)
    const float* memory = (const float*)d_in[1];  // (64,2048,512)
    const float* pm     = (const float*)d_in[2];  // (64,2048,128)
    const float* awc    = (const float*)d_in[3];  // (64,2,2048)
    const float* Wq     = (const float*)d_in[4];  // (128,1024)
    const float* Wv     = (const float*)d_in[5];  // (128)
    const float* convw  = (const float*)d_in[6];  // (32,2,31)
    const float* convb  = (const float*)d_in[7];  // (32)
    const float* Wloc   = (const float*)d_in[8];  // (128,32)

    float* ctx = (float*)d_out;                   // (64,512)
    float* aw  = (float*)d_out + B_ * EMB_;       // (64,2048)

    float* ws       = (float*)d_ws;
    float* pq       = ws;                                   // 64*128
    float* energies = pq + B_ * ADIM_;                      // 64*2048
    float* partial  = energies + (long)B_ * T_;             // 64*NCH*512

    pq_kernel<<<4, 256, 0, stream>>>(query, Wq, pq);
    energies_kernel<<<dim3(T_ / TT, B_), 256, 0, stream>>>(
        awc, pm, convw, convb, Wloc, Wv, pq, energies);
    softmax_kernel<<<B_, 256, 0, stream>>>(energies, aw);
    context_partial_kernel<<<dim3(NCH, B_), 256, 0, stream>>>(aw, memory, partial);
    context_reduce_kernel<<<B_, 256, 0, stream>>>(partial, ctx);
}